// StackedRNNEncoder_46462956208395
// MI455X (gfx1250) — compile-verified
//
#include <hip/hip_runtime.h>
#include <hip/hip_bf16.h>

// ---------------------------------------------------------------------------
// StackedRNNEncoder for MI455X (gfx1250, wave32, WMMA, big-LDS WGP)
// B=64, T=512, I=H=512, 2 layers.
//   Kernel 1: Z0 = x @ Wx0 + bx0      (parallel fp32 WMMA GEMM, 4 N-tiles/wave)
//   Kernel 2: persistent 8-WG recurrence; h state staged in LDS (padded rows),
//             fp32 WMMA, atomic grid barrier per phase
//   Kernel 3: gather last = outputs[b, len[b]-1]
// Z0 lives in the outputs region of d_out and is overwritten in-place by the
// layer-1 hidden states (read in phase A, written in phase B, same t,
// separated by a grid barrier).
// ---------------------------------------------------------------------------

#define BB   64
#define TT   512
#define HH   512
#define BT   (BB * TT)        // 32768
#define NWG  8                // workgroups in persistent recurrent kernel
#define HSTATE (BB * HH)      // 32768 floats per hidden-state buffer
#define PAD  516              // LDS row stride in floats (516 % 64 == 4 -> no bank conflicts)
#define LDS_BYTES (2u * 64u * PAD * 4u)   // two staged 64x512 A matrices (264192 B)

typedef float v2f __attribute__((ext_vector_type(2)));
typedef float v8f __attribute__((ext_vector_type(8)));

// D = A(16x4,f32) * B(4x16,f32) + C(16x16,f32)
__device__ __forceinline__ v8f wmma_f32_k4(v2f a, v2f b, v8f c) {
  return __builtin_amdgcn_wmma_f32_16x16x4_f32(
      /*neg_a=*/false, a, /*neg_b=*/false, b,
      /*c_mod=*/(short)0, c, /*reuse_a=*/false, /*reuse_b=*/false);
}

// ---------------------------------------------------------------------------
// Device-scope grid barrier for the persistent kernel (NWG co-resident WGs).
// bar[0] = arrive counter, bar[1] = generation.
// ---------------------------------------------------------------------------
__device__ __forceinline__ void grid_barrier(unsigned* bar) {
  __syncthreads();
  if (threadIdx.x == 0) {
    __threadfence();
    unsigned gen = __hip_atomic_load(&bar[1], __ATOMIC_RELAXED,
                                     __HIP_MEMORY_SCOPE_AGENT);
    unsigned prev = __hip_atomic_fetch_add(&bar[0], 1u, __ATOMIC_ACQ_REL,
                                           __HIP_MEMORY_SCOPE_AGENT);
    if (prev == NWG - 1) {
      __hip_atomic_store(&bar[0], 0u, __ATOMIC_RELAXED,
                         __HIP_MEMORY_SCOPE_AGENT);
      __hip_atomic_store(&bar[1], gen + 1u, __ATOMIC_RELEASE,
                         __HIP_MEMORY_SCOPE_AGENT);
    } else {
      while (__hip_atomic_load(&bar[1], __ATOMIC_ACQUIRE,
                               __HIP_MEMORY_SCOPE_AGENT) == gen) {
        __builtin_amdgcn_s_sleep(2);
      }
    }
    __threadfence();
  }
  __syncthreads();
}

// Cooperative stage of a 64x512 row-major matrix into LDS with padded rows.
// 256 threads, float4 (b128) transfers; dst rows are 16B aligned (PAD*4=2064).
__device__ __forceinline__ void stage_h_to_lds(const float* __restrict__ src,
                                               float* __restrict__ dst) {
#pragma unroll
  for (int it = 0; it < (HSTATE / 4) / 256; ++it) {
    const int idx = it * 256 + threadIdx.x;   // float4 index
    const int row = idx >> 7;                 // 128 float4 per row
    const int c4 = idx & 127;
    const float4 v = ((const float4*)src)[idx];
    float* d = dst + row * PAD + c4 * 4;
    d[0] = v.x; d[1] = v.y; d[2] = v.z; d[3] = v.w;
  }
}

// ---------------------------------------------------------------------------
// Kernel 1: Z[m][n] = sum_k X[m][k] * W[k][n] + bias[n]
//   X: (BT x HH), W: (HH x HH), Z: (BT x HH), all row-major.
//   Each wave computes a 16x64 output block (4 N-tiles sharing each A load).
// ---------------------------------------------------------------------------
__global__ void __launch_bounds__(256)
rnn_in_gemm_kernel(const float* __restrict__ X, const float* __restrict__ W,
                   const float* __restrict__ bias, float* __restrict__ Z) {
  const int wave = blockIdx.x * (blockDim.x >> 5) + (threadIdx.x >> 5);
  const int ngroups = (HH / 16) / 4;       // 8 column groups of 4 tiles
  const int mt = wave / ngroups;           // 0..2047 row tile
  const int ng = wave % ngroups;           // 0..7
  const int lane = threadIdx.x & 31;
  const int half = lane >> 4;
  const int l15 = lane & 15;
  const int col0 = ng * 64 + l15;          // lane's column in tile j: col0 + j*16

  const float* __restrict__ arow = X + (size_t)(mt * 16 + l15) * HH;

  v8f acc[4] = {};
#pragma unroll 2
  for (int k0 = 0; k0 < HH; k0 += 4) {
    const int ka = k0 + 2 * half;          // ISA A layout: VGPR0 K=ka, VGPR1 K=ka+1
    v2f a = {arow[ka], arow[ka + 1]};
    const float* __restrict__ w0 = W + (size_t)ka * HH + col0;
    const float* __restrict__ w1 = W + (size_t)(ka + 1) * HH + col0;
#pragma unroll
    for (int j = 0; j < 4; ++j) {
      v2f b = {w0[j * 16], w1[j * 16]};
      acc[j] = wmma_f32_k4(a, b, acc[j]);
    }
  }

#pragma unroll
  for (int j = 0; j < 4; ++j) {
    const int col = col0 + j * 16;
    const float bn = bias[col];
#pragma unroll
    for (int r = 0; r < 8; ++r) {
      const int m = mt * 16 + r + 8 * half;  // C/D layout: VGPR r -> M=r (+8 hi lanes)
      Z[(size_t)m * HH + col] = acc[j][r] + bn;
    }
  }
}

// ---------------------------------------------------------------------------
// Kernel 2: persistent recurrence. NWG=8 WGs x 256 threads (8 waves).
//   WG w owns output columns [w*64, w*64+64). Wave v: column tile (v>>1),
//   rows (v&1)*32 .. +32 (two 16x16 accumulators sharing each B tile).
// A-operands (h state) staged in LDS with padded rows; weights from L2.
// Per step t:
//   phase A: h0' = mask(tanh(Z0[t] + h0 @ Wh0 + bh0))          [grid barrier]
//   phase B: h1' = mask(tanh(h0' @ Wx1 + h1 @ Wh1 + bx1+bh1)),
//            outputs[t] = h1'                                   [grid barrier]
// ---------------------------------------------------------------------------
__global__ void __launch_bounds__(256)
rnn_recurrent_kernel(const float* __restrict__ Wh0, const float* __restrict__ bh0,
                     const float* __restrict__ Wx1, const float* __restrict__ bx1,
                     const float* __restrict__ Wh1, const float* __restrict__ bh1,
                     const int* __restrict__ lengths,
                     float* __restrict__ ZO,        // (BT x HH): Z0 in, outputs out
                     float* __restrict__ h0buf,     // 2 * HSTATE (parity buffers)
                     float* __restrict__ h1buf,     // 2 * HSTATE
                     unsigned* __restrict__ bar) {
  extern __shared__ float lds[];
  float* __restrict__ ldsA = lds;              // 64 x PAD
  float* __restrict__ ldsB = lds + 64 * PAD;   // 64 x PAD

  const int wg = blockIdx.x;               // 0..7
  const int wv = threadIdx.x >> 5;         // 0..7
  const int lane = threadIdx.x & 31;
  const int half = lane >> 4;
  const int l15 = lane & 15;
  const int ncol = wg * 64 + (wv >> 1) * 16 + l15;  // lane's output column
  const int mbase = (wv & 1) * 32;                   // rows mbase..mbase+31

  const float vbh0 = bh0[ncol];
  const float vb1 = bx1[ncol] + bh1[ncol];

  const int rowA0 = (mbase + l15) * PAD;        // LDS row offsets for A tiles
  const int rowA1 = (mbase + 16 + l15) * PAD;

  for (int t = 0; t < TT; ++t) {
    const int par = t & 1;
    const float* __restrict__ h0_old = h0buf + par * HSTATE;
    float* __restrict__ h0_new       = h0buf + (par ^ 1) * HSTATE;
    const float* __restrict__ h1_old = h1buf + par * HSTATE;
    float* __restrict__ h1_new       = h1buf + (par ^ 1) * HSTATE;

    // ---------------- phase A: h0' = mask(tanh(Z0[t] + h0 @ Wh0 + bh0)) ----
    stage_h_to_lds(h0_old, ldsA);
    __syncthreads();

    v8f acc0 = {}, acc1 = {};
#pragma unroll 4
    for (int k0 = 0; k0 < HH; k0 += 4) {
      const int ka = k0 + 2 * half;
      v2f b  = {Wh0[(size_t)ka * HH + ncol], Wh0[(size_t)(ka + 1) * HH + ncol]};
      v2f a0 = {ldsA[rowA0 + ka], ldsA[rowA0 + ka + 1]};
      v2f a1 = {ldsA[rowA1 + ka], ldsA[rowA1 + ka + 1]};
      acc0 = wmma_f32_k4(a0, b, acc0);
      acc1 = wmma_f32_k4(a1, b, acc1);
    }
#pragma unroll
    for (int tile = 0; tile < 2; ++tile) {
      const v8f acc = tile ? acc1 : acc0;
#pragma unroll
      for (int r = 0; r < 8; ++r) {
        const int m = mbase + tile * 16 + r + 8 * half;   // batch row
        const float z = ZO[((size_t)m * TT + t) * HH + ncol];
        const float v = tanhf(acc[r] + z + vbh0);
        const float hv = (lengths[m] > t) ? v : ldsA[m * PAD + ncol];
        h0_new[(size_t)m * HH + ncol] = hv;
      }
    }
    grid_barrier(bar);

    // ---------------- phase B: h1' = mask(tanh(h0'@Wx1 + h1@Wh1 + b)) ------
    stage_h_to_lds(h0_new, ldsA);   // full h0' (all WGs' slices, post-barrier)
    stage_h_to_lds(h1_old, ldsB);
    __syncthreads();

    acc0 = (v8f){};
    acc1 = (v8f){};
#pragma unroll 2
    for (int k0 = 0; k0 < HH; k0 += 4) {
      const int ka = k0 + 2 * half;
      v2f bx  = {Wx1[(size_t)ka * HH + ncol], Wx1[(size_t)(ka + 1) * HH + ncol]};
      v2f bh  = {Wh1[(size_t)ka * HH + ncol], Wh1[(size_t)(ka + 1) * HH + ncol]};
      v2f a0x = {ldsA[rowA0 + ka], ldsA[rowA0 + ka + 1]};
      v2f a1x = {ldsA[rowA1 + ka], ldsA[rowA1 + ka + 1]};
      v2f a0h = {ldsB[rowA0 + ka], ldsB[rowA0 + ka + 1]};
      v2f a1h = {ldsB[rowA1 + ka], ldsB[rowA1 + ka + 1]};
      acc0 = wmma_f32_k4(a0x, bx, acc0);
      acc1 = wmma_f32_k4(a1x, bx, acc1);
      acc0 = wmma_f32_k4(a0h, bh, acc0);
      acc1 = wmma_f32_k4(a1h, bh, acc1);
    }
#pragma unroll
    for (int tile = 0; tile < 2; ++tile) {
      const v8f acc = tile ? acc1 : acc0;
#pragma unroll
      for (int r = 0; r < 8; ++r) {
        const int m = mbase + tile * 16 + r + 8 * half;
        const float v = tanhf(acc[r] + vb1);
        const float hv = (lengths[m] > t) ? v : ldsB[m * PAD + ncol];
        h1_new[(size_t)m * HH + ncol] = hv;
        ZO[((size_t)m * TT + t) * HH + ncol] = hv;   // outputs[b, t, n]
      }
    }
    grid_barrier(bar);
  }
}

// ---------------------------------------------------------------------------
// Kernel 3: last[b, n] = outputs[b, lengths[b]-1, n]
// ---------------------------------------------------------------------------
__global__ void __launch_bounds__(256)
rnn_last_kernel(const float* __restrict__ outputs,
                const int* __restrict__ lengths, float* __restrict__ last) {
  const int i = blockIdx.x * blockDim.x + threadIdx.x;
  if (i >= BB * HH) return;
  const int b = i >> 9;        // / HH
  const int n = i & (HH - 1);
  const int tl = lengths[b] - 1;
  last[i] = outputs[((size_t)b * TT + tl) * HH + n];
}

// ---------------------------------------------------------------------------
// Init: zero hidden-state double buffers and barrier state.
// ---------------------------------------------------------------------------
__global__ void __launch_bounds__(256)
rnn_init_kernel(float* __restrict__ h0buf, float* __restrict__ h1buf,
                unsigned* __restrict__ bar) {
  const int i = blockIdx.x * blockDim.x + threadIdx.x;
  if (i < 2 * HSTATE) {
    h0buf[i] = 0.0f;
    h1buf[i] = 0.0f;
  }
  if (i < 2) bar[i] = 0u;
}

// ---------------------------------------------------------------------------
extern "C" void kernel_launch(void* const* d_in, const int* in_sizes, int n_in,
                              void* d_out, int out_size, void* d_ws,
                              size_t ws_size, hipStream_t stream) {
  (void)in_sizes; (void)n_in; (void)out_size; (void)ws_size;

  const float* x       = (const float*)d_in[0];  // (B, T, I)
  const int*   lengths = (const int*)d_in[1];    // (B,)
  const float* Wx0 = (const float*)d_in[2];
  const float* bx0 = (const float*)d_in[3];
  const float* Wh0 = (const float*)d_in[4];
  const float* bh0 = (const float*)d_in[5];
  const float* Wx1 = (const float*)d_in[6];
  const float* bx1 = (const float*)d_in[7];
  const float* Wh1 = (const float*)d_in[8];
  const float* bh1 = (const float*)d_in[9];

  float* outputs = (float*)d_out;                        // (B, T, H)
  float* last    = outputs + (size_t)BB * TT * HH;       // (B, H)

  float*    h0buf = (float*)d_ws;                        // 2 * HSTATE
  float*    h1buf = h0buf + 2 * HSTATE;                  // 2 * HSTATE
  unsigned* bar   = (unsigned*)(h1buf + 2 * HSTATE);     // 2 words

  // 1) zero state + barrier
  rnn_init_kernel<<<(2 * HSTATE + 255) / 256, 256, 0, stream>>>(h0buf, h1buf,
                                                                bar);
  // 2) Z0 = x @ Wx0 + bx0, staged into the outputs region of d_out
  //    2048 row-tiles x 8 column-groups = 16384 waves -> 2048 WGs
  rnn_in_gemm_kernel<<<(BT / 16) * ((HH / 16) / 4) / 8, 256, 0, stream>>>(
      x, Wx0, bx0, outputs);
  // 3) persistent recurrence over T steps (264 KB dynamic LDS per WG)
  rnn_recurrent_kernel<<<NWG, 256, LDS_BYTES, stream>>>(
      Wh0, bh0, Wx1, bx1, Wh1, bh1, lengths, outputs, h0buf, h1buf, bar);
  // 4) gather last hidden states
  rnn_last_kernel<<<(BB * HH + 255) / 256, 256, 0, stream>>>(outputs, lengths,
                                                             last);
}